// DeformBlock_71107478552929
// MI455X (gfx1250) — compile-verified
//
#include <hip/hip_runtime.h>

// Problem constants (match reference)
#define BB   4
#define CC   256
#define HH   128
#define WW   128
#define HWSZ (HH * WW)          // 16384
#define BNC  64
#define K9   9
#define KK   (BNC * K9)         // 576
#define NPIX (BB * HWSZ)        // 65536
#define MT   (NPIX / 16)        // 4096 M-tiles of 16 pixels

typedef __attribute__((ext_vector_type(2))) float v2f;
typedef __attribute__((ext_vector_type(8))) float v8f;

__device__ __forceinline__ v8f wmma4(v2f a, v2f b, v8f c) {
    // D = A(16x4 f32) * B(4x16 f32) + C(16x16 f32), full fp32 matrix op
    return __builtin_amdgcn_wmma_f32_16x16x4_f32(
        /*neg_a=*/false, a, /*neg_b=*/false, b,
        /*c_mod=*/(short)0, c, /*reuse_a=*/false, /*reuse_b=*/false);
}

__device__ __forceinline__ float sigmoidf_(float x) {
    return 1.0f / (1.0f + __expf(-x));
}

// ---------------- Stage 1: 1x1 down projection (M=NPIX, N=64, K=256) --------
// One wave per 16-pixel M-tile; 4 accumulators cover all of N=64, so each
// A fragment (x) is loaded once and feeds 4 WMMAs.
__global__ __launch_bounds__(128) void k_down(const float* __restrict__ x,
                                              const float* __restrict__ dw,
                                              float* __restrict__ z) {
    int Mtile = (blockIdx.x * blockDim.x + threadIdx.x) >> 5;   // 4096 waves
    int lane  = threadIdx.x & 31;
    int lm = lane & 15, half = lane >> 4;

    int p  = Mtile * 16 + lm;
    int b  = p >> 14, hw = p & (HWSZ - 1);
    const float* xb = x + (size_t)b * CC * HWSZ + hw;

    v8f acc[4] = {{}, {}, {}, {}};
    for (int k0 = 0; k0 < CC; k0 += 4) {
        int ka = k0 + 2 * half;
        v2f A;
        A.x = xb[(size_t)ka * HWSZ];
        A.y = xb[(size_t)(ka + 1) * HWSZ];
#pragma unroll
        for (int nt = 0; nt < 4; ++nt) {
            const float* wb = dw + (nt * 16 + lm) * CC;
            v2f Bf;
            Bf.x = wb[ka];
            Bf.y = wb[ka + 1];
            acc[nt] = wmma4(A, Bf, acc[nt]);
        }
    }
#pragma unroll
    for (int nt = 0; nt < 4; ++nt) {
        int n = nt * 16 + lm;
        const float* accf = (const float*)&acc[nt];
#pragma unroll
        for (int r = 0; r < 8; ++r) {
            int q = Mtile * 16 + r + 8 * half;
            int bq = q >> 14, hwq = q & (HWSZ - 1);
            z[((size_t)bq * BNC + n) * HWSZ + hwq] = accf[r];
        }
    }
}

// ---- Stage 2: fused offset(18)+mask(9) 3x3 conv as implicit-im2col GEMM ----
// K = 64*9 = 576, N padded 27->32; one wave per M-tile, 2 accumulators.
__global__ __launch_bounds__(128) void k_offmask(const float* __restrict__ z,
                                                 const float* __restrict__ offw,
                                                 const float* __restrict__ offb,
                                                 const float* __restrict__ maskw,
                                                 const float* __restrict__ maskb,
                                                 float* __restrict__ offo,
                                                 float* __restrict__ masko) {
    int Mtile = (blockIdx.x * blockDim.x + threadIdx.x) >> 5;   // 4096 waves
    int lane  = threadIdx.x & 31;
    int lm = lane & 15, half = lane >> 4;

    int p = Mtile * 16 + lm;
    int b = p >> 14, hw = p & (HWSZ - 1);
    int h = hw >> 7, w = hw & (WW - 1);
    const float* zb = z + (size_t)b * BNC * HWSZ;

    v8f acc[2] = {{}, {}};
    for (int k0 = 0; k0 < KK; k0 += 4) {
        v2f A;
#pragma unroll
        for (int t = 0; t < 2; ++t) {
            int k = k0 + 2 * half + t;
            int c = k / 9, j = k - c * 9;
            int yy = h + j / 3 - 1, xx = w + (j % 3) - 1;
            float av = 0.0f;
            if (yy >= 0 && yy < HH && xx >= 0 && xx < WW)
                av = zb[(size_t)c * HWSZ + yy * WW + xx];
            if (t == 0) A.x = av; else A.y = av;
        }
#pragma unroll
        for (int nt = 0; nt < 2; ++nt) {
            int n = nt * 16 + lm;
            v2f Bf;
#pragma unroll
            for (int t = 0; t < 2; ++t) {
                int k = k0 + 2 * half + t;
                float bv = 0.0f;
                if (n < 18)      bv = offw[n * KK + k];
                else if (n < 27) bv = maskw[(n - 18) * KK + k];
                if (t == 0) Bf.x = bv; else Bf.y = bv;
            }
            acc[nt] = wmma4(A, Bf, acc[nt]);
        }
    }
#pragma unroll
    for (int nt = 0; nt < 2; ++nt) {
        int n = nt * 16 + lm;
        const float* accf = (const float*)&acc[nt];
#pragma unroll
        for (int r = 0; r < 8; ++r) {
            int q = Mtile * 16 + r + 8 * half;
            int bq = q >> 14, hwq = q & (HWSZ - 1);
            if (n < 18)
                offo[((size_t)bq * 18 + n) * HWSZ + hwq] = accf[r] + offb[n];
            else if (n < 27)
                masko[((size_t)bq * 9 + (n - 18)) * HWSZ + hwq] =
                    sigmoidf_(accf[r] + maskb[n - 18]);
        }
    }
}

// ------ Stage 3: modulated deformable conv as fused gather + WMMA GEMM ------
// One wave per 16-pixel M-tile; 4 accumulators cover all of N=64, so the
// expensive modulated-bilinear A fragment is gathered ONCE per K-step and
// feeds 4 WMMAs. Per-wave bilinear tables (16 px x 9 taps) live in LDS.
__global__ __launch_bounds__(128) void k_deform(const float* __restrict__ z,
                                                const float* __restrict__ offo,
                                                const float* __restrict__ masko,
                                                const float* __restrict__ dfw,
                                                const float* __restrict__ dfb,
                                                float* __restrict__ zd) {
    __shared__ int   s_y0[4 * 144], s_x0[4 * 144];
    __shared__ float s_fy[4 * 144], s_fx[4 * 144], s_m[4 * 144];

    int wv    = threadIdx.x >> 5;                 // wave slot in block
    int lane  = threadIdx.x & 31;
    int Mtile = blockIdx.x * 4 + wv;              // one M-tile per wave
    int base  = wv * 144;

    for (int i = lane; i < 144; i += 32) {
        int pi = i / 9, j = i - pi * 9;
        int p = Mtile * 16 + pi;
        int b = p >> 14, hw = p & (HWSZ - 1);
        int h = hw >> 7, w = hw & (WW - 1);
        float dy = offo[((size_t)b * 18 + 2 * j) * HWSZ + hw];
        float dx = offo[((size_t)b * 18 + 2 * j + 1) * HWSZ + hw];
        float py = dy + (float)(h + j / 3 - 1);
        float px = dx + (float)(w + (j % 3) - 1);
        float fy0 = floorf(py), fx0 = floorf(px);
        s_y0[base + i] = (int)fy0;  s_x0[base + i] = (int)fx0;
        s_fy[base + i] = py - fy0;  s_fx[base + i] = px - fx0;
        s_m[base + i]  = masko[((size_t)b * 9 + j) * HWSZ + hw];
    }
    __syncthreads();

    int lm = lane & 15, half = lane >> 4;
    int p = Mtile * 16 + lm;
    int b = p >> 14;
    const float* zb = z + (size_t)b * BNC * HWSZ;

    v8f acc[4] = {{}, {}, {}, {}};
    for (int k0 = 0; k0 < KK; k0 += 4) {
        v2f A;
#pragma unroll
        for (int t = 0; t < 2; ++t) {
            int k = k0 + 2 * half + t;
            int c = k / 9, j = k - c * 9;
            int e = base + lm * 9 + j;
            int y0 = s_y0[e], x0 = s_x0[e];
            float fy = s_fy[e], fx = s_fx[e];
            const float* zc = zb + (size_t)c * HWSZ;
            float s = 0.0f;
            bool yv0 = (y0 >= 0) & (y0 < HH);
            bool yv1 = (y0 + 1 >= 0) & (y0 + 1 < HH);
            bool xv0 = (x0 >= 0) & (x0 < WW);
            bool xv1 = (x0 + 1 >= 0) & (x0 + 1 < WW);
            if (yv0 & xv0) s += (1.0f - fy) * (1.0f - fx) * zc[y0 * WW + x0];
            if (yv0 & xv1) s += (1.0f - fy) * fx          * zc[y0 * WW + x0 + 1];
            if (yv1 & xv0) s += fy * (1.0f - fx)          * zc[(y0 + 1) * WW + x0];
            if (yv1 & xv1) s += fy * fx                   * zc[(y0 + 1) * WW + x0 + 1];
            s *= s_m[e];
            if (t == 0) A.x = s; else A.y = s;
        }
#pragma unroll
        for (int nt = 0; nt < 4; ++nt) {
            const float* wrow = dfw + (size_t)(nt * 16 + lm) * KK;
            v2f Bf;
            Bf.x = wrow[k0 + 2 * half];
            Bf.y = wrow[k0 + 2 * half + 1];
            acc[nt] = wmma4(A, Bf, acc[nt]);
        }
    }
#pragma unroll
    for (int nt = 0; nt < 4; ++nt) {
        int n = nt * 16 + lm;
        const float* accf = (const float*)&acc[nt];
#pragma unroll
        for (int r = 0; r < 8; ++r) {
            int q = Mtile * 16 + r + 8 * half;
            int bq = q >> 14, hwq = q & (HWSZ - 1);
            zd[((size_t)bq * BNC + n) * HWSZ + hwq] = accf[r] + dfb[n];
        }
    }
}

// -------- Stage 4: 1x1 up projection (K=64, N=256) fused with residual ------
// One wave per (M-tile, N-group of 64); 4 accumulators share each A load.
__global__ __launch_bounds__(128) void k_up(const float* __restrict__ zd,
                                            const float* __restrict__ uw,
                                            const float* __restrict__ x,
                                            float* __restrict__ y) {
    int wvg  = (blockIdx.x * blockDim.x + threadIdx.x) >> 5;    // 16384 waves
    int lane = threadIdx.x & 31;
    int lm = lane & 15, half = lane >> 4;
    int Mtile = wvg >> 2, grp = wvg & 3;          // grp selects 64 of N=256

    int p = Mtile * 16 + lm;
    int b = p >> 14, hw = p & (HWSZ - 1);
    const float* zb = zd + (size_t)b * BNC * HWSZ + hw;

    v8f acc[4] = {{}, {}, {}, {}};
    for (int k0 = 0; k0 < BNC; k0 += 4) {
        int ka = k0 + 2 * half;
        v2f A;
        A.x = zb[(size_t)ka * HWSZ];
        A.y = zb[(size_t)(ka + 1) * HWSZ];
#pragma unroll
        for (int nt = 0; nt < 4; ++nt) {
            int n = (grp * 4 + nt) * 16 + lm;
            const float* wrow = uw + n * BNC;
            v2f Bf;
            Bf.x = wrow[ka];
            Bf.y = wrow[ka + 1];
            acc[nt] = wmma4(A, Bf, acc[nt]);
        }
    }
#pragma unroll
    for (int nt = 0; nt < 4; ++nt) {
        int n = (grp * 4 + nt) * 16 + lm;
        const float* accf = (const float*)&acc[nt];
#pragma unroll
        for (int r = 0; r < 8; ++r) {
            int q = Mtile * 16 + r + 8 * half;
            int bq = q >> 14, hwq = q & (HWSZ - 1);
            size_t idx = ((size_t)bq * CC + n) * HWSZ + hwq;
            y[idx] = accf[r] + x[idx];
        }
    }
}

// ----------- Stage 5: per-channel batch statistics (mean, rsqrt var) --------
__global__ __launch_bounds__(256) void k_stats(const float* __restrict__ y,
                                               float* __restrict__ st) {
    __shared__ float ss[256], sq[256];
    int c = blockIdx.x;
    float s = 0.0f, q = 0.0f;
    for (int b = 0; b < BB; ++b) {
        const float* yb = y + ((size_t)b * CC + c) * HWSZ;
        for (int i = threadIdx.x; i < HWSZ; i += 256) {
            float v = yb[i];
            s += v; q += v * v;
        }
    }
    ss[threadIdx.x] = s; sq[threadIdx.x] = q;
    __syncthreads();
    for (int o = 128; o > 0; o >>= 1) {
        if (threadIdx.x < o) {
            ss[threadIdx.x] += ss[threadIdx.x + o];
            sq[threadIdx.x] += sq[threadIdx.x + o];
        }
        __syncthreads();
    }
    if (threadIdx.x == 0) {
        const float invN = 1.0f / (float)(BB * HWSZ);
        float mean = ss[0] * invN;
        float var  = sq[0] * invN - mean * mean;
        st[c]      = mean;
        st[CC + c] = rsqrtf(var + 1e-5f);
    }
}

// ---------------- Stage 6: BatchNorm affine + SiLU, in place on d_out -------
__global__ __launch_bounds__(256) void k_bnsilu(float* __restrict__ y,
                                                const float* __restrict__ st,
                                                const float* __restrict__ gamma,
                                                const float* __restrict__ beta) {
    size_t idx = (size_t)blockIdx.x * blockDim.x + threadIdx.x;
    if (idx >= (size_t)BB * CC * HWSZ) return;
    int c = (int)((idx >> 14) & (CC - 1));
    float v  = y[idx];
    float yh = (v - st[c]) * st[CC + c] * gamma[c] + beta[c];
    y[idx] = yh * sigmoidf_(yh);
}

extern "C" void kernel_launch(void* const* d_in, const int* in_sizes, int n_in,
                              void* d_out, int out_size, void* d_ws, size_t ws_size,
                              hipStream_t stream) {
    const float* x      = (const float*)d_in[0];
    const float* down_w = (const float*)d_in[1];
    const float* off_w  = (const float*)d_in[2];
    const float* off_b  = (const float*)d_in[3];
    const float* mask_w = (const float*)d_in[4];
    const float* mask_b = (const float*)d_in[5];
    const float* def_w  = (const float*)d_in[6];
    const float* def_b  = (const float*)d_in[7];
    const float* up_w   = (const float*)d_in[8];
    const float* gamma  = (const float*)d_in[9];
    const float* beta   = (const float*)d_in[10];
    float* out = (float*)d_out;

    // Workspace partition (~40.6 MB of fp32) — all intermediates L2-resident.
    float* ws    = (float*)d_ws;
    float* z     = ws;                                  // [4][64][16384]
    float* offo  = z    + (size_t)BB * BNC * HWSZ;      // [4][18][16384]
    float* masko = offo + (size_t)BB * 18  * HWSZ;      // [4][9][16384]
    float* zd    = masko + (size_t)BB * 9  * HWSZ;      // [4][64][16384]
    float* st    = zd   + (size_t)BB * BNC * HWSZ;      // [2][256]

    // 4096 waves, 4 per block
    k_down<<<MT / 4, 128, 0, stream>>>(x, down_w, z);
    // 4096 waves, 4 per block
    k_offmask<<<MT / 4, 128, 0, stream>>>(z, off_w, off_b, mask_w, mask_b,
                                          offo, masko);
    // 4096 waves (one M-tile each), 4 per block with per-wave LDS tables
    k_deform<<<MT / 4, 128, 0, stream>>>(z, offo, masko, def_w, def_b, zd);
    // 16384 waves (M-tile x 4 N-groups), 4 per block
    k_up<<<MT * 4 / 4, 128, 0, stream>>>(zd, up_w, x, out);
    k_stats<<<CC, 256, 0, stream>>>(out, st);
    size_t total = (size_t)BB * CC * HWSZ;
    k_bnsilu<<<(int)((total + 255) / 256), 256, 0, stream>>>(out, st, gamma, beta);
}